// SharedPointSetAttention_29832842838757
// MI455X (gfx1250) — compile-verified
//
#include <hip/hip_runtime.h>
#include <math.h>

typedef __attribute__((ext_vector_type(2))) float v2f;
typedef __attribute__((ext_vector_type(8))) float v8f;

#define NNODES 30000
#define CH     128
#define NEDGE  480000
#define BN_EPS 1e-5f

// ---------- helpers ----------
__device__ __forceinline__ unsigned f2ord(float f) {
  unsigned u = __float_as_uint(f);
  return (u & 0x80000000u) ? ~u : (u | 0x80000000u);
}
__device__ __forceinline__ float ord2f(unsigned u) {
  return (u & 0x80000000u) ? __uint_as_float(u ^ 0x80000000u)
                           : __uint_as_float(~u);
}

__global__ void zero_f32(float* __restrict__ p, long n) {
  long i = (long)blockIdx.x * blockDim.x + threadIdx.x;
  if (i < n) p[i] = 0.0f;
}

// ---------- dense: Y[r, colOfs+c] = act((X @ W^T + bias) * scale + shift) ----------
// X: [n, K] row-major (stride K); W: [128, K] row-major.
// block = (32, 8): 8 waves, each wave computes one 16x16 tile of the 128 output cols.
// Uses V_WMMA_F32_16X16X4_F32 (full fp32) -- dense FLOPs are tiny vs the sparse side,
// and the softmax downstream is sensitive to logit precision.
__global__ void linear_wmma(const float* __restrict__ X,
                            const float* __restrict__ W,
                            const float* __restrict__ bias,
                            const float* __restrict__ gamma,  // null -> plain linear
                            const float* __restrict__ beta,
                            float* __restrict__ Y,
                            int K, int ldy, int colOfs, int doRelu) {
  const int lane = threadIdx.x;        // 0..31
  const int wave = threadIdx.y;        // 0..7  -> column tile
  const int half = lane >> 4;          // 0/1
  const int m    = lane & 15;
  const int rowBase = blockIdx.x * 16;
  const int colBase = wave * 16;

  // A 16x4 f32 layout: lane holds A[M=m][kbase + 2*half + {0,1}]
  // B 4x16 f32 layout: lane holds B[kbase + 2*half + {0,1}][N=m] = W[colBase+m][...]
  const float* xrow = X + (long)(rowBase + m) * K + 2 * half;
  const float* wrow = W + (long)(colBase + m) * K + 2 * half;

  v8f acc = {};
#pragma unroll 8
  for (int k = 0; k < K; k += 4) {
    v2f a = *(const v2f*)(xrow + k);
    v2f b = *(const v2f*)(wrow + k);
    acc = __builtin_amdgcn_wmma_f32_16x16x4_f32(
        /*neg_a=*/false, a, /*neg_b=*/false, b,
        /*c_mod=*/(short)0, acc, /*reuse_a=*/false, /*reuse_b=*/false);
  }

  const int ocol = colBase + m;
  float sc, sh;
  if (gamma) {  // Linear -> eval BN folded: y = (xW^T+b)*g/sqrt(1+eps) + beta
    sc = gamma[ocol] * rsqrtf(1.0f + BN_EPS);
    sh = bias[ocol] * sc + beta[ocol];
  } else {
    sc = 1.0f;
    sh = bias[ocol];
  }
  // D layout: VGPR v holds D[M = v + 8*half][N = m]
#pragma unroll
  for (int v = 0; v < 8; ++v) {
    float y = acc[v] * sc + sh;
    if (doRelu) y = fmaxf(y, 0.0f);
    Y[(long)(rowBase + v + 8 * half) * ldy + colOfs + ocol] = y;
  }
}

// ---------- sparse phase 1: per-edge dot + segment max ----------
// wave per edge; lane covers 4 channels (float4), wave-reduce, lane0 commits.
__global__ void edge_logits_max(const float* __restrict__ Kf,
                                const float* __restrict__ Qf,
                                const int* __restrict__ g0,
                                const int* __restrict__ g1,
                                float* __restrict__ logits,
                                unsigned* __restrict__ menc,
                                int nEdges) {
  int e = blockIdx.x * 8 + (threadIdx.x >> 5);
  int lane = threadIdx.x & 31;
  if (e >= nEdges) return;
  int d = g0[e], s = g1[e];
  const float4 kv = *(const float4*)(Kf + (long)s * CH + lane * 4);
  const float4 qv = *(const float4*)(Qf + (long)d * CH + lane * 4);
  float p = kv.x * qv.x + kv.y * qv.y + kv.z * qv.z + kv.w * qv.w;
#pragma unroll
  for (int off = 16; off > 0; off >>= 1) p += __shfl_down(p, off, 32);
  if (lane == 0) {
    logits[e] = p;
    atomicMax(&menc[d], f2ord(p));
  }
}

// ---------- sparse phase 2: e = exp(logit - m[seg]); segment sum ----------
__global__ void edge_exp_sum(const int* __restrict__ g0,
                             const unsigned* __restrict__ menc,
                             float* __restrict__ logits,   // in: logits, out: exp
                             float* __restrict__ ssum,
                             int nEdges) {
  int e = blockIdx.x * blockDim.x + threadIdx.x;
  if (e >= nEdges) return;
  int d = g0[e];
  unsigned enc = menc[d];
  float mval = (enc == 0u) ? 0.0f : ord2f(enc);   // empty segment -> 0 (isfinite branch)
  float ev = expf(logits[e] - mval);
  logits[e] = ev;
  atomicAdd(&ssum[d], ev);
}

// ---------- sparse phase 3: out[seg] += (e/s[seg]) * v[seg_row] ----------
__global__ void edge_scatter(const float* __restrict__ Vf,
                             const int* __restrict__ g0,
                             const float* __restrict__ expv,
                             const float* __restrict__ ssum,
                             float* __restrict__ Out, int ldOut, int colOfs,
                             int nEdges) {
  int e = blockIdx.x * 8 + (threadIdx.x >> 5);
  int lane = threadIdx.x & 31;
  if (e >= nEdges) return;
  int d = g0[e];
  float w = expv[e] / ssum[d];
  const float4 vv = *(const float4*)(Vf + (long)d * CH + lane * 4);
  float* o = Out + (long)d * ldOut + colOfs + lane * 4;
  atomicAdd(o + 0, w * vv.x);
  atomicAdd(o + 1, w * vv.y);
  atomicAdd(o + 2, w * vv.z);
  atomicAdd(o + 3, w * vv.w);
}

// ---------- host ----------
extern "C" void kernel_launch(void* const* d_in, const int* in_sizes, int n_in,
                              void* d_out, int out_size, void* d_ws, size_t ws_size,
                              hipStream_t stream) {
  (void)in_sizes; (void)n_in; (void)out_size; (void)ws_size;
  const float* feat1 = (const float*)d_in[0];
  const float* feat2 = (const float*)d_in[2];
  // q/k branches: w, b, g, beta  for q1, k1, q2, k2
  const float* wq1 = (const float*)d_in[4];  const float* bq1 = (const float*)d_in[5];
  const float* gq1 = (const float*)d_in[6];  const float* betaq1 = (const float*)d_in[7];
  const float* wk1 = (const float*)d_in[8];  const float* bk1 = (const float*)d_in[9];
  const float* gk1 = (const float*)d_in[10]; const float* betak1 = (const float*)d_in[11];
  const float* wq2 = (const float*)d_in[12]; const float* bq2 = (const float*)d_in[13];
  const float* gq2 = (const float*)d_in[14]; const float* betaq2 = (const float*)d_in[15];
  const float* wk2 = (const float*)d_in[16]; const float* bk2 = (const float*)d_in[17];
  const float* gk2 = (const float*)d_in[18]; const float* betak2 = (const float*)d_in[19];
  const float* wv1 = (const float*)d_in[20]; const float* bv1 = (const float*)d_in[21];
  const float* wv2 = (const float*)d_in[22]; const float* bv2 = (const float*)d_in[23];
  const float* wp1 = (const float*)d_in[24]; const float* bp1 = (const float*)d_in[25];
  const float* wp2 = (const float*)d_in[26]; const float* bp2 = (const float*)d_in[27];
  const int* graph1  = (const int*)d_in[28];
  const int* graph2  = (const int*)d_in[29];
  const int* graph12 = (const int*)d_in[30];
  const int* graph21 = (const int*)d_in[31];

  const long NC = (long)NNODES * CH;
  float* ws = (float*)d_ws;
  float* q1 = ws;
  float* k1 = q1 + NC;
  float* v1 = k1 + NC;
  float* q2 = v1 + NC;
  float* k2 = q2 + NC;
  float* v2 = k2 + NC;
  float* cat1 = v2 + NC;                       // [N, 256]
  float* cat2 = cat1 + (long)NNODES * 256;     // [N, 256]
  float* mencf = cat2 + (long)NNODES * 256;    // 4*N encoded max
  unsigned* menc = (unsigned*)mencf;
  float* ssum = mencf + 4L * NNODES;           // 4*N segment sums
  float* lgs  = ssum + 4L * NNODES;            // 4*E logits/exp

  // Zero all accumulators (cat1, cat2, menc, ssum are contiguous).
  {
    long zn = 2L * NNODES * 256 + 8L * NNODES;
    zero_f32<<<(unsigned)((zn + 255) / 256), 256, 0, stream>>>(cat1, zn);
  }

  dim3 gb(32, 8);
  const int rowTiles = NNODES / 16;  // 1875

  // q/k (Linear+BN+ReLU) and v (Linear) for both sets
  linear_wmma<<<rowTiles, gb, 0, stream>>>(feat1, wq1, bq1, gq1, betaq1, q1, CH, CH, 0, 1);
  linear_wmma<<<rowTiles, gb, 0, stream>>>(feat1, wk1, bk1, gk1, betak1, k1, CH, CH, 0, 1);
  linear_wmma<<<rowTiles, gb, 0, stream>>>(feat1, wv1, bv1, nullptr, nullptr, v1, CH, CH, 0, 0);
  linear_wmma<<<rowTiles, gb, 0, stream>>>(feat2, wq2, bq2, gq2, betaq2, q2, CH, CH, 0, 1);
  linear_wmma<<<rowTiles, gb, 0, stream>>>(feat2, wk2, bk2, gk2, betak2, k2, CH, CH, 0, 1);
  linear_wmma<<<rowTiles, gb, 0, stream>>>(feat2, wv2, bv2, nullptr, nullptr, v2, CH, CH, 0, 0);

  const int eWaveBlocks = NEDGE / 8;            // 60000 (exact)
  const int eThrBlocks  = (NEDGE + 255) / 256;

  // attn(key, query, value, graph) -> Out[:, colOfs:colOfs+128]
  auto attn = [&](const float* Kf, const float* Qf, const float* Vf, const int* graph,
                  unsigned* m, float* s, float* lg, float* Out, int colOfs) {
    const int* g0 = graph;            // graph[0]
    const int* g1 = graph + NEDGE;    // graph[1]
    edge_logits_max<<<eWaveBlocks, 256, 0, stream>>>(Kf, Qf, g0, g1, lg, m, NEDGE);
    edge_exp_sum<<<eThrBlocks, 256, 0, stream>>>(g0, m, lg, s, NEDGE);
    edge_scatter<<<eWaveBlocks, 256, 0, stream>>>(Vf, g0, lg, s, Out, 256, colOfs, NEDGE);
  };

  attn(k1, q1, v1, graph1,  menc + 0L * NNODES, ssum + 0L * NNODES, lgs + 0L * NEDGE, cat1, 0);   // feat11
  attn(k2, q2, v2, graph2,  menc + 1L * NNODES, ssum + 1L * NNODES, lgs + 1L * NEDGE, cat2, 0);   // feat22
  attn(k2, q1, v1, graph21, menc + 2L * NNODES, ssum + 2L * NNODES, lgs + 2L * NEDGE, cat1, 128); // feat12
  attn(k1, q2, v2, graph12, menc + 3L * NNODES, ssum + 3L * NNODES, lgs + 3L * NEDGE, cat2, 128); // feat21

  // Output projections: [N,256] @ [128,256]^T + b  ->  d_out
  float* out = (float*)d_out;
  linear_wmma<<<rowTiles, gb, 0, stream>>>(cat1, wp1, bp1, nullptr, nullptr, out,            256, CH, 0, 0);
  linear_wmma<<<rowTiles, gb, 0, stream>>>(cat2, wp2, bp2, nullptr, nullptr, out + NC,       256, CH, 0, 0);
}